// OnnxRoiAlign_7919919693913
// MI455X (gfx1250) — compile-verified
//
#include <hip/hip_runtime.h>
#include <hip/hip_bf16.h>

// -----------------------------------------------------------------------------
// ROI Align forward for MI455X (gfx1250, wave32).
//
// Shape (fixed by reference):
//   feat: (8, 256, 128, 128) f32   rois: (1000, 4) f32   bidx: (1000,) i32
//   out : (1000, 256, 7, 7) f32, out_h=out_w=7, sampling_ratio=2, scale=0.25
//
// Roofline: feat = 128MB (fully L2-resident, 192MB L2), out = 50MB streaming,
// ~0.5 GFLOP total -> pure gather-bandwidth/latency problem. Strategy:
//   * 256-thread block lies within exactly one ROI (C*7*7 = 12544 = 49*256): ROI box +
//     batch index are block-uniform -> scalar SMEM loads + SALU math.
//   * Branchless: all 16 corner gathers per output are unconditional
//     (coords pre-clamped, validity folded into a multiplicative mask).
//   * CDNA5 async gather: all 16 corners issued as
//     GLOBAL_LOAD_ASYNC_TO_LDS_B32 (ASYNCcnt-tracked, no VGPR return slots
//     held), one S_WAIT_ASYNCCNT 0, then conflict-free ds_load readback.
//     stage[16][256]: writes and reads are lane-consecutive -> 0 bank
//     conflicts; 16KB/block keeps LDS occupancy unconstrained (320KB/WGP).
//   * Non-temporal store for the streaming output (don't evict L2-hot feat).
// -----------------------------------------------------------------------------

namespace {
constexpr int   kOutHW  = 7;
constexpr int   kC      = 256;
constexpr int   kH      = 128;
constexpr int   kW      = 128;
constexpr float kScale  = 0.25f;
constexpr int   kBins   = kOutHW * kOutHW;   // 49
constexpr int   kPerRoi = kC * kBins;        // 12544 == 49 blocks of 256
}  // namespace

#if defined(__AMDGCN__) && __has_builtin(__builtin_amdgcn_global_load_async_to_lds_b32)
#define CDNA5_HAS_ASYNC_LDS 1
#endif

#if defined(CDNA5_HAS_ASYNC_LDS)
typedef __attribute__((address_space(1))) int gint_t;   // global
typedef __attribute__((address_space(3))) int lint_t;   // LDS

__device__ __forceinline__ void async_g2l_b32(const float* g, float* l) {
  __builtin_amdgcn_global_load_async_to_lds_b32((gint_t*)g, (lint_t*)l,
                                                /*offset=*/0, /*cpol=*/0);
}

__device__ __forceinline__ void wait_async0() {
#if __has_builtin(__builtin_amdgcn_s_wait_asynccnt)
  __builtin_amdgcn_s_wait_asynccnt(0);
#else
  asm volatile("s_wait_asynccnt 0" ::: "memory");
#endif
}
#endif  // CDNA5_HAS_ASYNC_LDS

__global__ __launch_bounds__(256) void roi_align_fwd(
    const float* __restrict__ feat, const float* __restrict__ rois,
    const int* __restrict__ bidx, float* __restrict__ out) {
  // Block-uniform ROI id (SGPR); per-lane bin/channel decomposition.
  const unsigned r     = blockIdx.x / kBins;
  const unsigned local = (blockIdx.x % kBins) * 256u + threadIdx.x;
  const unsigned pw    = local % kOutHW;
  const unsigned ph    = (local / kOutHW) % kOutHW;
  const unsigned c     = local / kBins;

  // ROI params: uniform per block -> scalar loads + SALU math.
  const float x1 = rois[r * 4 + 0] * kScale;
  const float y1 = rois[r * 4 + 1] * kScale;
  const float x2 = rois[r * 4 + 2] * kScale;
  const float y2 = rois[r * 4 + 3] * kScale;
  const float bin_w = fmaxf(x2 - x1, 1.0f) * (1.0f / kOutHW);
  const float bin_h = fmaxf(y2 - y1, 1.0f) * (1.0f / kOutHW);
  const int   n     = bidx[r];

  const float* __restrict__ fb =
      feat + ((size_t)((unsigned)n * kC + c)) * (size_t)(kH * kW);

  // Per-axis sample data (sampling_ratio = 2 -> offsets 0.25, 0.75).
  int   ylo[2], yhi[2];
  float ly[2], hy[2], my[2];
#pragma unroll
  for (int iy = 0; iy < 2; ++iy) {
    const float yy = y1 + ((float)ph + (iy ? 0.75f : 0.25f)) * bin_h;
    my[iy] = (yy >= -1.0f && yy <= (float)kH) ? 1.0f : 0.0f;
    const float yc = fminf(fmaxf(yy, 0.0f), (float)(kH - 1));
    const int   lo = (int)yc;           // yc >= 0, trunc == floor
    ylo[iy] = lo;
    yhi[iy] = min(lo + 1, kH - 1);
    ly[iy]  = yc - (float)lo;
    hy[iy]  = 1.0f - ly[iy];
  }
  int   xlo[2], xhi[2];
  float lx[2], hx[2], mx[2];
#pragma unroll
  for (int ix = 0; ix < 2; ++ix) {
    const float xx = x1 + ((float)pw + (ix ? 0.75f : 0.25f)) * bin_w;
    mx[ix] = (xx >= -1.0f && xx <= (float)kW) ? 1.0f : 0.0f;
    const float xc = fminf(fmaxf(xx, 0.0f), (float)(kW - 1));
    const int   lo = (int)xc;
    xlo[ix] = lo;
    xhi[ix] = min(lo + 1, kW - 1);
    lx[ix]  = xc - (float)lo;
    hx[ix]  = 1.0f - lx[ix];
  }

  // ---- gather 16 corner values ------------------------------------------
  float v[16];
#if defined(CDNA5_HAS_ASYNC_LDS)
  // CDNA5 path: 16 per-lane async gathers -> LDS, one ASYNCcnt wait,
  // conflict-free readback (each thread owns column threadIdx.x).
  __shared__ float stage[16][256];
#pragma unroll
  for (int iy = 0; iy < 2; ++iy) {
    const float* __restrict__ r0 = fb + ylo[iy] * kW;
    const float* __restrict__ r1 = fb + yhi[iy] * kW;
#pragma unroll
    for (int ix = 0; ix < 2; ++ix) {
      const int k = (iy * 2 + ix) * 4;
      async_g2l_b32(r0 + xlo[ix], &stage[k + 0][threadIdx.x]);
      async_g2l_b32(r0 + xhi[ix], &stage[k + 1][threadIdx.x]);
      async_g2l_b32(r1 + xlo[ix], &stage[k + 2][threadIdx.x]);
      async_g2l_b32(r1 + xhi[ix], &stage[k + 3][threadIdx.x]);
    }
  }
  wait_async0();  // per-wave; each thread reads only its own column
#pragma unroll
  for (int k = 0; k < 16; ++k) v[k] = stage[k][threadIdx.x];
#else
  // Fallback: direct VGPR-return gathers.
#pragma unroll
  for (int iy = 0; iy < 2; ++iy) {
    const float* __restrict__ r0 = fb + ylo[iy] * kW;
    const float* __restrict__ r1 = fb + yhi[iy] * kW;
#pragma unroll
    for (int ix = 0; ix < 2; ++ix) {
      const int k = (iy * 2 + ix) * 4;
      v[k + 0] = r0[xlo[ix]];
      v[k + 1] = r0[xhi[ix]];
      v[k + 2] = r1[xlo[ix]];
      v[k + 3] = r1[xhi[ix]];
    }
  }
#endif

  // ---- masked bilinear combine, mean over 2x2 samples -------------------
  float acc = 0.0f;
#pragma unroll
  for (int iy = 0; iy < 2; ++iy) {
#pragma unroll
    for (int ix = 0; ix < 2; ++ix) {
      const int   k   = (iy * 2 + ix) * 4;
      const float top = fmaf(lx[ix], v[k + 1], hx[ix] * v[k + 0]);
      const float bot = fmaf(lx[ix], v[k + 3], hx[ix] * v[k + 2]);
      const float bil = fmaf(ly[iy], bot, hy[iy] * top);
      acc = fmaf(my[iy] * mx[ix], bil, acc);
    }
  }

  __builtin_nontemporal_store(acc * 0.25f,
                              out + (size_t)r * kPerRoi + (size_t)local);
}

// -----------------------------------------------------------------------------
extern "C" void kernel_launch(void* const* d_in, const int* in_sizes, int n_in,
                              void* d_out, int out_size, void* d_ws,
                              size_t ws_size, hipStream_t stream) {
  const float* feat = (const float*)d_in[0];
  const float* rois = (const float*)d_in[1];
  const int*   bidx = (const int*)d_in[2];
  float*       out  = (float*)d_out;

  const int num_rois = out_size / kPerRoi;     // 1000
  const int blocks   = num_rois * kBins;       // 49000 blocks x 256 threads
  roi_align_fwd<<<blocks, 256, 0, stream>>>(feat, rois, bidx, out);
}